// TransEDecoder_67044439491159
// MI455X (gfx1250) — compile-verified
//
#include <hip/hip_runtime.h>

typedef __attribute__((ext_vector_type(2))) float v2f;
typedef __attribute__((ext_vector_type(8))) float v8f;

#define TRANSE_GAMMA 12.0f
#define EMB_D 256
#define WAVES_PER_BLOCK 8
#define TRIPLETS_PER_WAVE 16

// One wave32 handles 16 triplets.
// Lane L (L<16)  : triplet L, dims [  0..127] -> partials K=0 (dims 0..63),  K=1 (64..127)
// Lane L+16      : triplet L, dims [128..255] -> partials K=2 (128..191),    K=3 (192..255)
// This is exactly the V_WMMA_F32_16X16X4_F32 A-matrix layout (16x4 f32, 2 VGPRs).
// One WMMA vs an all-ones B folds the 4 partials per triplet:
//   D[m][n] = sum_k A[m][k] = L1 distance of triplet m (replicated over columns n).
// C/D layout: VGPR v, lanes 0-15 hold D[v][col], lanes 16-31 hold D[v+8][col],
// so lane 0 stores triplets base+0..7 and lane 16 stores base+8..15 — each as
// two contiguous global_store_b128s (output rows are consecutive).
__global__ __launch_bounds__(256) void transe_decoder_kernel(
    const float* __restrict__ node_emb,
    const float* __restrict__ rel_emb,
    const int*   __restrict__ triplets,
    float*       __restrict__ out,
    int B)
{
    const int lane      = threadIdx.x & 31;
    const int waveLocal = threadIdx.x >> 5;
    const int waveId    = blockIdx.x * WAVES_PER_BLOCK + waveLocal;
    const int base      = waveId * TRIPLETS_PER_WAVE;
    if (base >= B) return;                 // wave-uniform: EXEC stays all-ones

    const int m    = lane & 15;            // triplet row within the 16x16 tile
    const int half = lane >> 4;            // 0: dims 0..127, 1: dims 128..255

    int t = base + m;
    if (t > B - 1) t = B - 1;              // clamp tail loads (stores are guarded)

    const int hIdx = triplets[t * 3 + 0];
    const int rIdx = triplets[t * 3 + 1];
    const int tIdx = triplets[t * 3 + 2];

    const float4* hp = (const float4*)(node_emb + (size_t)hIdx * EMB_D) + half * 32;
    const float4* rp = (const float4*)(rel_emb  + (size_t)rIdx * EMB_D) + half * 32;
    const float4* tp = (const float4*)(node_emb + (size_t)tIdx * EMB_D) + half * 32;

    v2f a;
#pragma unroll
    for (int s = 0; s < 2; ++s) {          // two 64-dim K-slices per lane
        float acc = 0.0f;
#pragma unroll
        for (int i = 0; i < 16; ++i) {     // 16 x float4 = 64 dims
            const float4 hv = hp[s * 16 + i];
            const float4 rv = rp[s * 16 + i];
            const float4 tv = tp[s * 16 + i];
            acc += __builtin_fabsf(hv.x + rv.x - tv.x);
            acc += __builtin_fabsf(hv.y + rv.y - tv.y);
            acc += __builtin_fabsf(hv.z + rv.z - tv.z);
            acc += __builtin_fabsf(hv.w + rv.w - tv.w);
        }
        a[s] = acc;
    }

    v2f b;
    b[0] = 1.0f;
    b[1] = 1.0f;

    v8f c = {};
    // 8 args: (neg_a, A, neg_b, B, c_mod, C, reuse_a, reuse_b)
    c = __builtin_amdgcn_wmma_f32_16x16x4_f32(
        /*neg_a=*/false, a, /*neg_b=*/false, b,
        /*c_mod=*/(short)0, c, /*reuse_a=*/false, /*reuse_b=*/false);

    if (m == 0) {
        const int rowBase = base + half * 8;   // lane 0 -> rows 0..7, lane 16 -> rows 8..15
        if (rowBase + 8 <= B) {
            // Fast path (always taken when B % 16 == 0): two b128 stores, no exec churn.
            float4 lo, hi;
            lo.x = TRANSE_GAMMA - c[0];
            lo.y = TRANSE_GAMMA - c[1];
            lo.z = TRANSE_GAMMA - c[2];
            lo.w = TRANSE_GAMMA - c[3];
            hi.x = TRANSE_GAMMA - c[4];
            hi.y = TRANSE_GAMMA - c[5];
            hi.z = TRANSE_GAMMA - c[6];
            hi.w = TRANSE_GAMMA - c[7];
            *(float4*)(out + rowBase)     = lo;   // rowBase is 8-aligned -> 32B aligned
            *(float4*)(out + rowBase + 4) = hi;
        } else {
#pragma unroll
            for (int v = 0; v < 8; ++v) {
                const int o = rowBase + v;
                if (o < B) out[o] = TRANSE_GAMMA - c[v];
            }
        }
    }
}

extern "C" void kernel_launch(void* const* d_in, const int* in_sizes, int n_in,
                              void* d_out, int out_size, void* d_ws, size_t ws_size,
                              hipStream_t stream) {
    const float* node_emb = (const float*)d_in[0];
    const float* rel_emb  = (const float*)d_in[1];
    const int*   triplets = (const int*)d_in[2];
    float*       out      = (float*)d_out;

    const int B      = out_size;                                    // 500000 triplets
    const int waves  = (B + TRIPLETS_PER_WAVE - 1) / TRIPLETS_PER_WAVE;
    const int blocks = (waves + WAVES_PER_BLOCK - 1) / WAVES_PER_BLOCK;

    transe_decoder_kernel<<<blocks, WAVES_PER_BLOCK * 32, 0, stream>>>(
        node_emb, rel_emb, triplets, out, B);
}